// Resnet_with_skip_46042049413190
// MI455X (gfx1250) — compile-verified
//
#include <hip/hip_runtime.h>
#include <stdint.h>

#define AS1 __attribute__((address_space(1)))
#define AS3 __attribute__((address_space(3)))

typedef int v4i_t __attribute__((ext_vector_type(4)));

// ---- CDNA5 async global->LDS path (gfx1250), with safe fallback ----
#if __has_builtin(__builtin_amdgcn_global_load_async_to_lds_b128) && \
    __has_builtin(__builtin_amdgcn_global_load_async_to_lds_b32)
#define ASYNC_LDS 1
#endif

#if defined(ASYNC_LDS)
#  if __has_builtin(__builtin_amdgcn_s_wait_asynccnt)
#    define WAIT_ASYNC() __builtin_amdgcn_s_wait_asynccnt(0)
#  else
#    define WAIT_ASYNC() asm volatile("s_wait_asynccnt 0x0" ::: "memory")
#  endif
#else
#  define WAIT_ASYNC() ((void)0)
#endif

__device__ __forceinline__ void copy_g2l_b128(const float* g, float* l) {
#ifdef ASYNC_LDS
  // global src (AS1), LDS dst (AS3); low 32 bits of generic LDS ptr = LDS addr
  __builtin_amdgcn_global_load_async_to_lds_b128(
      (AS1 v4i_t*)(uintptr_t)g,
      (AS3 v4i_t*)(uint32_t)(uintptr_t)l, 0, 0);
#else
  *(float4*)l = *(const float4*)g;
#endif
}
__device__ __forceinline__ void copy_g2l_b32(const float* g, float* l) {
#ifdef ASYNC_LDS
  __builtin_amdgcn_global_load_async_to_lds_b32(
      (AS1 int*)(uintptr_t)g,
      (AS3 int*)(uint32_t)(uintptr_t)l, 0, 0);
#else
  *l = *g;
#endif
}

#define IMG_H 1024
#define IMG_W 1024
#define TILE  64

#define SIN_H 70      // tile + halo 3 each side
#define SIN_W 76      // 72 cols stored (aligned b128 chunks), stride padded (16B mult)
#define TMP_H 70      // horizontal-blur rows
#define TMP_W 68      // 66 cols used
#define BLR_H 66      // blurred: tile + halo 1
#define BLR_W 68      // 16B-multiple stride

// dy+1 / dx+1 for NMS direction k (2 bits each), k = octant = channel index
#define DYPACK 425     // dy = {0,1,1,1,0,-1,-1,-1}
#define DXPACK 36890   // dx = {1,1,0,-1,-1,-1,0,1}

__global__ __launch_bounds__(256)
void fused_edge_nms_kernel(const float* __restrict__ reconst,
                           const float* __restrict__ gauss_w,
                           const float* __restrict__ nms_w,
                           float* __restrict__ out)
{
  __shared__ __align__(16) float sIn[SIN_H * SIN_W];
  __shared__ __align__(16) float sTmp[TMP_H * TMP_W];
  __shared__ __align__(16) float sBlur[BLR_H * BLR_W];
  __shared__ float sVal[8];

  const int tid = threadIdx.x;
  const int x0  = blockIdx.x * TILE;
  const int y0  = blockIdx.y * TILE;
  const float* img  = reconst + (size_t)blockIdx.z * (IMG_H * IMG_W);
  float*       outp = out     + (size_t)blockIdx.z * (IMG_H * IMG_W);

  // ---- Stage A: issue async tile+halo load (zero-padded outside image) ----
  // Region: rows gy = y0-3 .. y0+66 (70), cols gx = x0-4 .. x0+67 (72, 16B aligned)
  for (int t = tid; t < SIN_H * 18; t += 256) {
    int r   = t / 18;
    int ch  = t - r * 18;
    int gy  = y0 - 3 + r;
    int gxc = x0 - 4 + ch * 4;
    float* ldst = &sIn[r * SIN_W + ch * 4];
    if ((unsigned)gy < (unsigned)IMG_H && gxc >= 0 && gxc + 3 < IMG_W) {
      copy_g2l_b128(&img[(size_t)gy * IMG_W + gxc], ldst);
    } else {
#pragma unroll
      for (int e = 0; e < 4; ++e) {
        int gx = gxc + e;
        if ((unsigned)gy < (unsigned)IMG_H && (unsigned)gx < (unsigned)IMG_W)
          copy_g2l_b32(&img[(size_t)gy * IMG_W + gx], ldst + e);
        else
          ldst[e] = 0.0f;  // disjoint from async targets -> no ordering hazard
      }
    }
  }

  // ---- Overlap: scalar weight prep while async copies fly ----
  // NMS directional weights: val[c] = nms_w[c,0,r_c,cc_c]
  if (tid < 8) {
    int r = (DYPACK >> (2 * tid)) & 3;   // r_c = dy+1
    int c = (DXPACK >> (2 * tid)) & 3;   // cc_c = dx+1
    sVal[tid] = nms_w[tid * 9 + r * 3 + c];
  }
  // Separable Gaussian: g = rowsum(g) (x) colsum(g) exactly
  float wr[5], wc[5];
#pragma unroll
  for (int i = 0; i < 5; ++i) { wr[i] = 0.0f; wc[i] = 0.0f; }
#pragma unroll
  for (int r = 0; r < 5; ++r)
#pragma unroll
    for (int c = 0; c < 5; ++c) {
      float g = gauss_w[r * 5 + c];
      wr[r] += g; wc[c] += g;
    }

  WAIT_ASYNC();
  __syncthreads();

  // ---- Stage B: horizontal blur (rows 0..69, cols 0..65 -> gx = x0-1..x0+64) ----
  for (int t = tid; t < TMP_H * 66; t += 256) {
    int r = t / 66;
    int c = t - r * 66;
    const float* s = &sIn[r * SIN_W + c + 1];
    sTmp[r * TMP_W + c] =
        wc[0] * s[0] + wc[1] * s[1] + wc[2] * s[2] + wc[3] * s[3] + wc[4] * s[4];
  }
  __syncthreads();

  // ---- Stage C: vertical blur -> blurred tile + halo 1 ----
  for (int t = tid; t < BLR_H * 66; t += 256) {
    int r = t / 66;
    int c = t - r * 66;
    const float* s = &sTmp[r * TMP_W + c];
    sBlur[r * BLR_W + c] =
        wr[0] * s[0] + wr[1] * s[TMP_W] + wr[2] * s[2 * TMP_W] +
        wr[3] * s[3 * TMP_W] + wr[4] * s[4 * TMP_W];
  }
  __syncthreads();

  // ---- Stage D: sobel + octant + NMS gather + threshold + gamma ----
  // 4 consecutive pixels per iteration; one b128 store per quad.
  const float TAN225 = 0.41421356237309503f;
  for (int t = tid; t < TILE * (TILE / 4); t += 256) {   // 1024 quad-tasks
    int py = t >> 4;
    int qx = (t & 15) << 2;
    int gy = y0 + py;
    int gx0 = x0 + qx;

    // y-clamp folded into row base pointers (replicate padding for Sobel)
    int yc0 = max(gy - 1, 0) - (y0 - 1);
    int yc1 = py + 1;
    int yc2 = min(gy + 1, IMG_H - 1) - (y0 - 1);
    const float* r0 = &sBlur[yc0 * BLR_W];
    const float* r1 = &sBlur[yc1 * BLR_W];
    const float* r2 = &sBlur[yc2 * BLR_W];

    float a0[8], a1[8], a2[8];
    const bool xfast = (gx0 > 0) && (gx0 + 4 < IMG_W);
    if (xfast) {
      // local cols qx .. qx+7 cover px-1 .. px+4 for all 4 pixels (use 6 of 8)
      *(float4*)&a0[0] = *(const float4*)&r0[qx];
      *(float4*)&a0[4] = *(const float4*)&r0[qx + 4];
      *(float4*)&a1[0] = *(const float4*)&r1[qx];
      *(float4*)&a1[4] = *(const float4*)&r1[qx + 4];
      *(float4*)&a2[0] = *(const float4*)&r2[qx];
      *(float4*)&a2[4] = *(const float4*)&r2[qx + 4];
    }

    float4 rc4 = *(const float4*)&sIn[(py + 3) * SIN_W + (qx + 4)];
    float res[4];

#pragma unroll
    for (int e = 0; e < 4; ++e) {
      int px = qx + e;
      int gx = gx0 + e;

      float n00, n01, n02, n10, b0, n12, n20, n21, n22;
      if (xfast) {
        n00 = a0[e]; n01 = a0[e + 1]; n02 = a0[e + 2];
        n10 = a1[e]; b0  = a1[e + 1]; n12 = a1[e + 2];
        n20 = a2[e]; n21 = a2[e + 1]; n22 = a2[e + 2];
      } else {
        int xc0 = max(gx - 1, 0) - (x0 - 1);
        int xc1 = px + 1;
        int xc2 = min(gx + 1, IMG_W - 1) - (x0 - 1);
        n00 = r0[xc0]; n01 = r0[xc1]; n02 = r0[xc2];
        n10 = r1[xc0]; b0  = r1[xc1]; n12 = r1[xc2];
        n20 = r2[xc0]; n21 = r2[xc1]; n22 = r2[xc2];
      }

      float gxs = (n02 - n00) + 2.0f * (n12 - n10) + (n22 - n20);
      float gys = (n20 - n00) + 2.0f * (n21 - n01) + (n22 - n02);

      // octant k == round(atan2(gys,gxs)/45deg) mod 8 ; <= on boundaries == half-to-even
      float ax = fabsf(gxs), ay = fabsf(gys);
      int k;
      if (ay <= TAN225 * ax)      k = (gxs >= 0.0f) ? 0 : 4;
      else if (ax <= TAN225 * ay) k = (gys >= 0.0f) ? 2 : 6;
      else if (gys >= 0.0f)       k = (gxs >= 0.0f) ? 1 : 3;
      else                        k = (gxs >= 0.0f) ? 7 : 5;

      int dy = ((DYPACK >> (2 * k)) & 3) - 1;
      int dx = ((DXPACK >> (2 * k)) & 3) - 1;
      int k2 = k ^ 4;  // (k+4) mod 8 ; opposite direction = (-dy,-dx)

      // NMS conv uses zero padding at image borders
      float v1 = 0.0f, v2 = 0.0f;
      if ((unsigned)(gy + dy) < (unsigned)IMG_H && (unsigned)(gx + dx) < (unsigned)IMG_W)
        v1 = sBlur[(py + 1 + dy) * BLR_W + (px + 1 + dx)];
      if ((unsigned)(gy - dy) < (unsigned)IMG_H && (unsigned)(gx - dx) < (unsigned)IMG_W)
        v2 = sBlur[(py + 1 - dy) * BLR_W + (px + 1 - dx)];

      float cp = fmaf(sVal[k],  v1, b0);
      float cn = fmaf(sVal[k2], v2, b0);
      float m  = fminf(cp, cn);
      m = (m > 0.01f) ? m : 0.01f;

      float rc = (e == 0) ? rc4.x : (e == 1) ? rc4.y : (e == 2) ? rc4.z : rc4.w;
      float o  = rc * m;
      res[e] = o * o;
    }

    float4 o4;
    o4.x = res[0]; o4.y = res[1]; o4.z = res[2]; o4.w = res[3];
    *(float4*)&outp[(size_t)gy * IMG_W + gx0] = o4;
  }
}

extern "C" void kernel_launch(void* const* d_in, const int* in_sizes, int n_in,
                              void* d_out, int out_size, void* d_ws, size_t ws_size,
                              hipStream_t stream) {
  const float* reconst = (const float*)d_in[0];
  const float* gauss_w = (const float*)d_in[1];
  const float* nms_w   = (const float*)d_in[2];
  float* out = (float*)d_out;

  int B = in_sizes[0] / (IMG_H * IMG_W);
  dim3 grid(IMG_W / TILE, IMG_H / TILE, B);
  fused_edge_nms_kernel<<<grid, 256, 0, stream>>>(reconst, gauss_w, nms_w, out);
}